// PartitionTCA_2044404433395
// MI455X (gfx1250) — compile-verified
//
#include <hip/hip_runtime.h>

// PartitionTCA construct(): out[i,j,k] = sum_r v0a[r,i]*v0b[r,j,k]
//                                      + sum_r v1a[r,j]*v1b[r,i,k]
//                                      + sum_r v2a[r,k]*v2b[r,i,j]
// D0=256 (i), D1=512 (j), D2=256 (k), ranks 8/8/8, all f32.
//
// One wave owns a 16(j) x 16(k) output tile and loops over i.
// Terms 2 and 3 are 16x16x8 f32 matmuls -> 4x v_wmma_f32_16x16x4_f32 per i.
// Term 1's tile (v0b[r, j-tile, k-tile]) is i-invariant: cached in 64 VGPRs,
// combined with 8 wave-uniform weights v0a[r,i] via VALU FMAs.
// Output written once with non-temporal stores (write-once 128MB stream;
// keep the ~10MB of reused factor tensors in L2).

typedef float v2f __attribute__((ext_vector_type(2)));
typedef float v8f __attribute__((ext_vector_type(8)));

#define TD0 256
#define TD1 512
#define TD2 256

__global__ __launch_bounds__(256) void ptca_fused_kernel(
    const float* __restrict__ v0a,  // [8][256]
    const float* __restrict__ v0b,  // [8][512][256]
    const float* __restrict__ v1a,  // [8][512]
    const float* __restrict__ v1b,  // [8][256][256]
    const float* __restrict__ v2a,  // [8][256]
    const float* __restrict__ v2b,  // [8][256][512]
    float* __restrict__ out)        // [256][512][256]
{
    const int lane = (int)(threadIdx.x & 31u);
    const int n    = lane & 15;        // N index within 16x16 tile (and A-matrix M index)
    const int hi   = lane >> 4;        // half-wave selector
    const int kq   = hi * 2;           // K-pair base inside a K=4 slab (A & B f32 layouts)

    // Wave-uniform task id: 8192 waves = 32 j-tiles x 16 k-tiles x 16 i-chunks.
    int task = __builtin_amdgcn_readfirstlane(
        (int)(blockIdx.x * 8u + (threadIdx.x >> 5)));
    const int ic  = task & 15;         // i-chunk (16 i values each)
    const int rem = task >> 4;
    const int kt  = rem & 15;          // k-tile
    const int jt  = rem >> 4;          // j-tile (0..31)
    const int i0  = ic * 16;
    const int k0  = kt * 16;
    const int j0  = jt * 16;

    // ---- term 2 A-matrix (loop-invariant): A2[m=j][r] = v1a[r][j0+m] ----
    // A 16x4 f32 layout: VGPR v, lane l -> A[l%16][2*(l>=16)+v]
    v2f a2_lo, a2_hi;
    a2_lo.x = v1a[(kq + 0) * TD1 + j0 + n];
    a2_lo.y = v1a[(kq + 1) * TD1 + j0 + n];
    a2_hi.x = v1a[(kq + 4) * TD1 + j0 + n];
    a2_hi.y = v1a[(kq + 5) * TD1 + j0 + n];

    // ---- term 3 B-matrix (loop-invariant): B3[r][n=k] = v2a[r][k0+n] ----
    // B 4x16 f32 layout (mirror of A): VGPR v, lane l -> B[2*(l>=16)+v][l%16]
    v2f b3_lo, b3_hi;
    b3_lo.x = v2a[(kq + 0) * TD2 + k0 + n];
    b3_lo.y = v2a[(kq + 1) * TD2 + k0 + n];
    b3_hi.x = v2a[(kq + 4) * TD2 + k0 + n];
    b3_hi.y = v2a[(kq + 5) * TD2 + k0 + n];

    // ---- term 1 tile cache (loop-invariant): c0[r][v] = v0b[r][j0+v+8*hi][k0+n]
    // Indexed exactly like the C/D 16x16 f32 layout (row = v + 8*hi, col = n).
    float c0[8][8];
#pragma unroll
    for (int r = 0; r < 8; ++r)
#pragma unroll
        for (int v = 0; v < 8; ++v)
            c0[r][v] = v0b[r * (TD1 * TD2) + (j0 + v + 8 * hi) * TD2 + k0 + n];

    for (int it = 0; it < 16; ++it) {
        const int ii = i0 + it;

        // term-1 weights: wave-uniform scalars v0a[r, ii]
        float w[8];
#pragma unroll
        for (int r = 0; r < 8; ++r) w[r] = v0a[r * TD0 + ii];

        // term-2 B-matrix: B2[r][n=k] = v1b[r][ii][k0+n]
        const int base2 = ii * TD2 + k0 + n;
        v2f b2_lo, b2_hi;
        b2_lo.x = v1b[(kq + 0) * (TD0 * TD2) + base2];
        b2_lo.y = v1b[(kq + 1) * (TD0 * TD2) + base2];
        b2_hi.x = v1b[(kq + 4) * (TD0 * TD2) + base2];
        b2_hi.y = v1b[(kq + 5) * (TD0 * TD2) + base2];

        // term-3 A-matrix: A3[m=j][r] = v2b[r][ii][j0+m]
        const int base3 = ii * TD1 + j0 + n;
        v2f a3_lo, a3_hi;
        a3_lo.x = v2b[(kq + 0) * (TD0 * TD1) + base3];
        a3_lo.y = v2b[(kq + 1) * (TD0 * TD1) + base3];
        a3_hi.x = v2b[(kq + 4) * (TD0 * TD1) + base3];
        a3_hi.y = v2b[(kq + 5) * (TD0 * TD1) + base3];

        // 16x16 f32 accumulator; chain all four K=4 WMMA slabs through C.
        v8f acc = {};
        acc = __builtin_amdgcn_wmma_f32_16x16x4_f32(
            false, a2_lo, false, b2_lo, (short)0, acc, false, false);
        acc = __builtin_amdgcn_wmma_f32_16x16x4_f32(
            false, a2_hi, false, b2_hi, (short)0, acc, false, false);
        acc = __builtin_amdgcn_wmma_f32_16x16x4_f32(
            false, a3_lo, false, b3_lo, (short)0, acc, false, false);
        acc = __builtin_amdgcn_wmma_f32_16x16x4_f32(
            false, a3_hi, false, b3_hi, (short)0, acc, false, false);

        // term 1: rank-8 weighted sum of the cached v0b tile (VALU, no memory).
#pragma unroll
        for (int v = 0; v < 8; ++v) {
            float s = acc[v];
#pragma unroll
            for (int r = 0; r < 8; ++r) s = fmaf(w[r], c0[r][v], s);
            acc[v] = s;
        }

        // Store the 16x16 tile: C/D layout row = v + 8*hi, col = n.
        float* op = out + ii * (TD1 * TD2) + (j0 + 8 * hi) * TD2 + k0 + n;
#pragma unroll
        for (int v = 0; v < 8; ++v)
            __builtin_nontemporal_store(acc[v], op + v * TD2);
    }
}

extern "C" void kernel_launch(void* const* d_in, const int* in_sizes, int n_in,
                              void* d_out, int out_size, void* d_ws, size_t ws_size,
                              hipStream_t stream) {
    const float* v0a = (const float*)d_in[0];  // [8][256]
    const float* v0b = (const float*)d_in[1];  // [8][512][256]
    const float* v1a = (const float*)d_in[2];  // [8][512]
    const float* v1b = (const float*)d_in[3];  // [8][256][256]
    const float* v2a = (const float*)d_in[4];  // [8][256]
    const float* v2b = (const float*)d_in[5];  // [8][256][512]
    float* out = (float*)d_out;                // [256][512][256]

    // 8192 waves = 1024 blocks x 8 waves; each wave: one (j,k) 16x16 tile,
    // 16 consecutive i values.
    dim3 grid(1024), block(256);
    hipLaunchKernelGGL(ptca_fused_kernel, grid, block, 0, stream,
                       v0a, v0b, v1a, v1b, v2a, v2b, out);
}